// ROCKET_33861522161790
// MI455X (gfx1250) — compile-verified
//
#include <hip/hip_runtime.h>
#include <math.h>

// ROCKET on MI455X (gfx1250):
//  - x [16,2,512] = 64KB staged once per block into LDS transposed [c][s][b]
//    via global_load_async_to_lds_b32 (ASYNCcnt), conflict-free reads.
//  - grid-stride over kernels: each 256-thread block (8 wave32) amortizes the
//    staging over several random kernels.
//  - thread = (b, 2 adjacent l) matching the V_WMMA_F32_16X16X4_F32 A layout;
//    PPV positive-count accumulated on the matrix unit (ones-matrix row-sum);
//    MAX in VALU + lane^16 shuffle + LDS reduce.
//  - zero-padding handled branch-free (clamped address + cndmask'd weight);
//    interior tiles take a fully unguarded dual-FMAC path (uniform s_branch).

#define C_IN  2
#define SEQ   512
#define BATCH 16
#define NKER  10000
#define OUTW  (2 * NKER)
#define XS_ELEMS (C_IN * SEQ * BATCH)   // 16384 floats = 64 KB
#define MAX_BLOCKS 1280

typedef float v2f __attribute__((ext_vector_type(2)));
typedef float v8f __attribute__((ext_vector_type(8)));

#if defined(__AMDGCN__) && __has_builtin(__builtin_amdgcn_wmma_f32_16x16x4_f32)
#define USE_WMMA 1
#else
#define USE_WMMA 0
#endif

#if defined(__AMDGCN__) && __has_builtin(__builtin_amdgcn_global_load_async_to_lds_b32)
#define USE_ASYNC 1
#else
#define USE_ASYNC 0
#endif

// Two convolution taps-sweeps for adjacent output positions l0, l0+1.
// CHECKED: clamp index + zero the weight (branch-free, no EXEC games).
template <int KS, bool CHECKED>
__device__ __forceinline__ void conv2(const float* __restrict__ xs,
                                      const float (&wr)[C_IN][KS],
                                      float bs, int qb, int d, int b,
                                      float& acc0, float& acc1) {
    acc0 = bs;
    acc1 = bs;
#pragma unroll
    for (int c = 0; c < C_IN; ++c) {
#pragma unroll
        for (int j = 0; j < KS; ++j) {
            const int q0 = qb + j * d;
            const int q1 = q0 + 1;
            const float wc = wr[c][j];
            if (CHECKED) {
                const bool in0 = (unsigned)q0 < (unsigned)SEQ;
                const bool in1 = (unsigned)q1 < (unsigned)SEQ;
                const int a0 = in0 ? q0 : 0;       // always-legal address
                const int a1 = in1 ? q1 : 0;
                const float v0 = xs[((c << 9) | a0) * BATCH + b];
                const float v1 = xs[((c << 9) | a1) * BATCH + b];
                acc0 = fmaf(in0 ? wc : 0.f, v0, acc0);  // zero-pad via weight
                acc1 = fmaf(in1 ? wc : 0.f, v1, acc1);
            } else {
                acc0 = fmaf(wc, xs[((c << 9) | q0) * BATCH + b], acc0);
                acc1 = fmaf(wc, xs[((c << 9) | q1) * BATCH + b], acc1);
            }
        }
    }
}

template <int KS>
__global__ __launch_bounds__(256, 2) void rocket_group_kernel(
    const float* __restrict__ x,     // [16,2,512]
    const float* __restrict__ w,     // [n, 2, KS]
    const float* __restrict__ bias,  // [n]
    const int* __restrict__ dil,     // [n]
    const int* __restrict__ pad,     // [n]  (p_layer in reference)
    const int* __restrict__ kidx,    // [n]
    const int* __restrict__ Larr,    // [n]
    float* __restrict__ out,         // [16, 20000]
    int nker)
{
    extern __shared__ float smem[];
    float* xs     = smem;               // [c][s][b], 16384 floats
    float* redMax = smem + XS_ELEMS;    // [8][16]
    float* redCnt = redMax + 128;       // [8][16]

    const int tid = threadIdx.x;

    // ---- Stage x into LDS once, transposed to [c][s][b] (per-lane scatter). ----
    for (int g = tid; g < XS_ELEMS; g += 256) {
        const int b = g >> 10;          // x is [b][c][s] row-major
        const int c = (g >> 9) & 1;
        const int s = g & (SEQ - 1);
        const int lo = (((c << 9) | s) * BATCH + b);
#if USE_ASYNC
        __builtin_amdgcn_global_load_async_to_lds_b32(
            (__attribute__((address_space(1))) int*)(void*)(x + g),
            (__attribute__((address_space(3))) int*)(xs + lo),
            0, 0);
#else
        xs[lo] = x[g];
#endif
    }
#if USE_ASYNC
#if __has_builtin(__builtin_amdgcn_s_wait_asynccnt)
    __builtin_amdgcn_s_wait_asynccnt(0);
#else
    asm volatile("s_wait_asynccnt 0" ::: "memory");
#endif
#endif
    __syncthreads();

    const int lane = tid & 31;
    const int wv   = tid >> 5;
    const int b    = lane & 15;          // WMMA A-matrix: lane -> M (=batch)
    const int k0   = (lane >> 4) << 1;   // lanes 0-15: K=0,1 ; lanes 16-31: K=2,3
#if USE_WMMA
    const v2f ones = {1.f, 1.f};
#endif

    // ---- Grid-stride over this ks-group's kernels. ----
    for (int n = blockIdx.x; n < nker; n += gridDim.x) {
        float wr[C_IN][KS];
#pragma unroll
        for (int c = 0; c < C_IN; ++c)
#pragma unroll
            for (int j = 0; j < KS; ++j)
                wr[c][j] = w[(n * C_IN + c) * KS + j];
        const float bs = bias[n];
        const int d = dil[n];
        const int p = pad[n];
        const int L = Larr[n];

        float mx = -__builtin_inff();
#if USE_WMMA
        v8f cnt = {0.f, 0.f, 0.f, 0.f, 0.f, 0.f, 0.f, 0.f};
#else
        float cntf = 0.f;
#endif

        // 8 waves x 4 l-positions = 32 output positions per block iteration.
        const int tiles = (L + 31) >> 5;   // uniform -> EXEC stays all-1s
        for (int t = 0; t < tiles; ++t) {
            const int lbase = t << 5;
            const int l0 = lbase + (wv << 2) + k0;   // this thread's l0, l0+1
            float acc0, acc1;
            // Whole-tile-in-signal test is block-uniform -> scalar branch.
            if (lbase >= p && (lbase + 31 - p + (KS - 1) * d + 1) < SEQ) {
                conv2<KS, false>(xs, wr, bs, l0 - p, d, b, acc0, acc1);
            } else {
                conv2<KS, true>(xs, wr, bs, l0 - p, d, b, acc0, acc1);
            }
            const bool v0 = (l0 < L);
            const bool v1 = (l0 + 1 < L);
            mx = fmaxf(mx, v0 ? acc0 : -__builtin_inff());
            mx = fmaxf(mx, v1 ? acc1 : -__builtin_inff());
#if USE_WMMA
            // Positive-count on the matrix engine: D = A(ind) x B(ones) + C
            v2f a;
            a.x = (v0 && acc0 > 0.f) ? 1.f : 0.f;
            a.y = (v1 && acc1 > 0.f) ? 1.f : 0.f;
            cnt = __builtin_amdgcn_wmma_f32_16x16x4_f32(
                false, a, false, ones, (short)0, cnt, false, false);
#else
            cntf += ((v0 && acc0 > 0.f) ? 1.f : 0.f) +
                    ((v1 && acc1 > 0.f) ? 1.f : 0.f);
#endif
        }

        // ---- Cross-wave reductions (WAR-protect red from previous iter). ----
        __syncthreads();
        mx = fmaxf(mx, __shfl_xor(mx, 16, 32));  // lane^16 has same b
        if (lane < 16) redMax[wv * 16 + b] = mx;
#if USE_WMMA
        // C/D layout: VGPR r -> M=r (lanes 0-15) / M=8+r (lanes 16-31);
        // every column of D is identical (B = ones).
        if (lane == 0) {
#pragma unroll
            for (int r = 0; r < 8; ++r) redCnt[wv * 16 + r] = cnt[r];
        } else if (lane == 16) {
#pragma unroll
            for (int r = 0; r < 8; ++r) redCnt[wv * 16 + 8 + r] = cnt[r];
        }
#else
        cntf += __shfl_xor(cntf, 16, 32);
        if (lane < 16) redCnt[wv * 16 + b] = cntf;
#endif
        __syncthreads();

        if (tid < 16) {  // tid = batch b
            float m = redMax[tid];
            float s = redCnt[tid];
#pragma unroll
            for (int w8 = 1; w8 < 8; ++w8) {
                m = fmaxf(m, redMax[w8 * 16 + tid]);
                s += redCnt[w8 * 16 + tid];
            }
            const int kid = kidx[n];
            float2 o;
            o.x = m;                 // max
            o.y = s / (float)L;      // ppv
            *(float2*)(out + tid * OUTW + 2 * kid) = o;  // global_store_b64
        }
    }
}

extern "C" void kernel_launch(void* const* d_in, const int* in_sizes, int n_in,
                              void* d_out, int out_size, void* d_ws, size_t ws_size,
                              hipStream_t stream) {
    // setup_inputs order:
    // 0:x | 1:w7 2:b7 3:dil7 4:pad7 5:kidx7 6:L7
    //     | 7:w9 8:b9 9:dil9 10:pad9 11:kidx9 12:L9
    //     | 13:w11 14:b11 15:dil11 16:pad11 17:kidx11 18:L11
    const float* x = (const float*)d_in[0];
    float* out = (float*)d_out;
    const int n7  = in_sizes[2];
    const int n9  = in_sizes[8];
    const int n11 = in_sizes[14];
    const size_t shmem = (XS_ELEMS + 256) * sizeof(float);  // 66,560 B

    if (n7 > 0) {
        const int g = n7 < MAX_BLOCKS ? n7 : MAX_BLOCKS;
        rocket_group_kernel<7><<<g, 256, shmem, stream>>>(
            x, (const float*)d_in[1], (const float*)d_in[2],
            (const int*)d_in[3], (const int*)d_in[4],
            (const int*)d_in[5], (const int*)d_in[6], out, n7);
    }
    if (n9 > 0) {
        const int g = n9 < MAX_BLOCKS ? n9 : MAX_BLOCKS;
        rocket_group_kernel<9><<<g, 256, shmem, stream>>>(
            x, (const float*)d_in[7], (const float*)d_in[8],
            (const int*)d_in[9], (const int*)d_in[10],
            (const int*)d_in[11], (const int*)d_in[12], out, n9);
    }
    if (n11 > 0) {
        const int g = n11 < MAX_BLOCKS ? n11 : MAX_BLOCKS;
        rocket_group_kernel<11><<<g, 256, shmem, stream>>>(
            x, (const float*)d_in[13], (const float*)d_in[14],
            (const int*)d_in[15], (const int*)d_in[16],
            (const int*)d_in[17], (const int*)d_in[18], out, n11);
    }
}